// DeepSets_62766652064048
// MI455X (gfx1250) — compile-verified
//
#include <hip/hip_runtime.h>

// DeepSets on MI455X (gfx1250): fp32 WMMA 16x16x4 for both MLPs,
// fp32 hardware atomics for the segment scatter.
//
// dims: N = 1,000,000 rows, D = 128 features, NSEG = 50,000 segments.

#define D     128
#define NSEG  50000
#define ROWS  32          // rows of X per block
#define ST    132         // LDS row stride (128 + 4 pad -> conflict-free A-frag reads)

typedef float v2f __attribute__((ext_vector_type(2)));
typedef float v8f __attribute__((ext_vector_type(8)));

__device__ __forceinline__ v8f wmma_f32(v2f a, v2f b, v8f c) {
    // D = A(16x4,f32) * B(4x16,f32) + C(16x16,f32)
    // 8-arg form: (neg_a, A, neg_b, B, c_mod, C, reuse_a, reuse_b)
    return __builtin_amdgcn_wmma_f32_16x16x4_f32(
        false, a, false, b, (short)0, c, false, false);
}

__global__ void zero_kernel(float* __restrict__ p, size_t n) {
    size_t i = (size_t)blockIdx.x * blockDim.x + threadIdx.x;
    size_t stride = (size_t)gridDim.x * blockDim.x;
    for (; i < n; i += stride) p[i] = 0.0f;
}

// ---------------------------------------------------------------------------
// phi MLP + scatter-add:  seg_sum[batch[i]] += relu(x W1 + b1) W2 + b2
// ---------------------------------------------------------------------------
__global__ __launch_bounds__(256)
void phi_scatter_kernel(const float* __restrict__ ins,
                        const int*   __restrict__ batch,
                        const float* __restrict__ W1, const float* __restrict__ b1,
                        const float* __restrict__ W2, const float* __restrict__ b2,
                        float* __restrict__ seg_sum, float* __restrict__ counts,
                        int N)
{
    __shared__ float Xs[ROWS * ST];
    __shared__ float Hs[ROWS * ST];

    const int tid  = threadIdx.x;
    const int wave = tid >> 5;        // 0..7  (wave32)
    const int lane = tid & 31;
    const int idx  = lane & 15;       // row (A) / col (B,D) within 16-tile
    const int half = lane >> 4;       // K-pair select for A/B frags
    const int base = blockIdx.x * ROWS;

    // --- stage X tile to LDS (coalesced float4) ---
    #pragma unroll
    for (int i = 0; i < 4; ++i) {
        int l   = tid + i * 256;      // 0..1023 float4 chunks
        int row = l >> 5;             // 32 float4 per row
        int c4  = l & 31;
        float4 v = make_float4(0.f, 0.f, 0.f, 0.f);
        if (base + row < N)
            v = ((const float4*)(ins + (size_t)(base + row) * D))[c4];
        *(float4*)&Xs[row * ST + c4 * 4] = v;
    }
    if (tid < ROWS && base + tid < N) {
        unsafeAtomicAdd(&counts[batch[base + tid]], 1.0f);
    }
    __syncthreads();

    const int nt  = wave;             // each wave owns one 16-wide column tile
    const int col = nt * 16 + idx;
    const float bias1 = b1[col];

    // --- layer 1: H = relu(X @ W1 + b1) ---
    #pragma unroll
    for (int mt = 0; mt < 2; ++mt) {
        v8f c = {};
        const float* xrow = &Xs[(mt * 16 + idx) * ST];
        #pragma unroll
        for (int k0 = 0; k0 < D; k0 += 4) {
            v2f a, b;
            a.x = xrow[k0 + 2 * half];
            a.y = xrow[k0 + 2 * half + 1];
            b.x = W1[(size_t)(k0 + 2 * half) * D + col];
            b.y = W1[(size_t)(k0 + 2 * half + 1) * D + col];
            c = wmma_f32(a, b, c);
        }
        #pragma unroll
        for (int r = 0; r < 8; ++r) {
            float h = c[r] + bias1;
            Hs[(mt * 16 + r + 8 * half) * ST + col] = h > 0.0f ? h : 0.0f;
        }
    }
    __syncthreads();

    // --- layer 2: Y = H @ W2 + b2, scatter-add into seg_sum ---
    const float bias2 = b2[col];
    #pragma unroll
    for (int mt = 0; mt < 2; ++mt) {
        v8f c = {};
        const float* hrow = &Hs[(mt * 16 + idx) * ST];
        #pragma unroll
        for (int k0 = 0; k0 < D; k0 += 4) {
            v2f a, b;
            a.x = hrow[k0 + 2 * half];
            a.y = hrow[k0 + 2 * half + 1];
            b.x = W2[(size_t)(k0 + 2 * half) * D + col];
            b.y = W2[(size_t)(k0 + 2 * half + 1) * D + col];
            c = wmma_f32(a, b, c);
        }
        #pragma unroll
        for (int r = 0; r < 8; ++r) {
            int row = base + mt * 16 + r + 8 * half;
            if (row < N) {
                int seg = batch[row];
                unsafeAtomicAdd(&seg_sum[(size_t)seg * D + col], c[r] + bias2);
            }
        }
    }
}

// ---------------------------------------------------------------------------
// rho MLP on segment means:  out = relu(mean W1 + b1) W2 + b2
// ---------------------------------------------------------------------------
__global__ __launch_bounds__(256)
void rho_kernel(const float* __restrict__ seg_sum, const float* __restrict__ counts,
                const float* __restrict__ W1, const float* __restrict__ b1,
                const float* __restrict__ W2, const float* __restrict__ b2,
                float* __restrict__ out)
{
    __shared__ float Xs[ROWS * ST];
    __shared__ float Hs[ROWS * ST];

    const int tid  = threadIdx.x;
    const int wave = tid >> 5;
    const int lane = tid & 31;
    const int idx  = lane & 15;
    const int half = lane >> 4;
    const int base = blockIdx.x * ROWS;

    // --- stage mean tile to LDS: seg_sum / max(count,1), zero for tail rows ---
    #pragma unroll
    for (int i = 0; i < 16; ++i) {
        int l    = tid + i * 256;     // 0..4095 elements (32 rows x 128)
        int row  = l >> 7;
        int ccol = l & 127;
        int seg  = base + row;
        float v = 0.0f;
        if (seg < NSEG) {
            float cnt = counts[seg];
            v = seg_sum[(size_t)seg * D + ccol] / fmaxf(cnt, 1.0f);
        }
        Xs[row * ST + ccol] = v;
    }
    __syncthreads();

    const int nt  = wave;
    const int col = nt * 16 + idx;
    const float bias1 = b1[col];

    #pragma unroll
    for (int mt = 0; mt < 2; ++mt) {
        v8f c = {};
        const float* xrow = &Xs[(mt * 16 + idx) * ST];
        #pragma unroll
        for (int k0 = 0; k0 < D; k0 += 4) {
            v2f a, b;
            a.x = xrow[k0 + 2 * half];
            a.y = xrow[k0 + 2 * half + 1];
            b.x = W1[(size_t)(k0 + 2 * half) * D + col];
            b.y = W1[(size_t)(k0 + 2 * half + 1) * D + col];
            c = wmma_f32(a, b, c);
        }
        #pragma unroll
        for (int r = 0; r < 8; ++r) {
            float h = c[r] + bias1;
            Hs[(mt * 16 + r + 8 * half) * ST + col] = h > 0.0f ? h : 0.0f;
        }
    }
    __syncthreads();

    const float bias2 = b2[col];
    #pragma unroll
    for (int mt = 0; mt < 2; ++mt) {
        v8f c = {};
        const float* hrow = &Hs[(mt * 16 + idx) * ST];
        #pragma unroll
        for (int k0 = 0; k0 < D; k0 += 4) {
            v2f a, b;
            a.x = hrow[k0 + 2 * half];
            a.y = hrow[k0 + 2 * half + 1];
            b.x = W2[(size_t)(k0 + 2 * half) * D + col];
            b.y = W2[(size_t)(k0 + 2 * half + 1) * D + col];
            c = wmma_f32(a, b, c);
        }
        #pragma unroll
        for (int r = 0; r < 8; ++r) {
            int seg = base + mt * 16 + r + 8 * half;
            if (seg < NSEG)
                out[(size_t)seg * D + col] = c[r] + bias2;
        }
    }
}

extern "C" void kernel_launch(void* const* d_in, const int* in_sizes, int n_in,
                              void* d_out, int out_size, void* d_ws, size_t ws_size,
                              hipStream_t stream)
{
    const float* ins  = (const float*)d_in[0];
    const int*   bat  = (const int*)  d_in[1];
    // d_in[2] = dim (scalar, always 0) -> ignored
    const float* pW1  = (const float*)d_in[3];
    const float* pb1  = (const float*)d_in[4];
    const float* pW2  = (const float*)d_in[5];
    const float* pb2  = (const float*)d_in[6];
    const float* rW1  = (const float*)d_in[7];
    const float* rb1  = (const float*)d_in[8];
    const float* rW2  = (const float*)d_in[9];
    const float* rb2  = (const float*)d_in[10];
    float* outp = (float*)d_out;

    const int N = in_sizes[0] / D;          // 1,000,000

    float* seg_sum = (float*)d_ws;          // [NSEG * D]
    float* counts  = seg_sum + (size_t)NSEG * D;  // [NSEG]

    zero_kernel<<<2048, 256, 0, stream>>>(seg_sum, (size_t)NSEG * (D + 1));

    int phi_blocks = (N + ROWS - 1) / ROWS;  // 31,250
    phi_scatter_kernel<<<phi_blocks, 256, 0, stream>>>(
        ins, bat, pW1, pb1, pW2, pb2, seg_sum, counts, N);

    int rho_blocks = (NSEG + ROWS - 1) / ROWS;  // 1,563
    rho_kernel<<<rho_blocks, 256, 0, stream>>>(
        seg_sum, counts, rW1, rb1, rW2, rb2, outp);
}